// GNN_40793599377789
// MI455X (gfx1250) — compile-verified
//
#include <hip/hip_runtime.h>

#define NNODES 100000
#define NEDGES 1600000
#define NGRAPH 128
#define FIN    768
#define HD     64
#define BN_EPS 1e-5f
#define ATT_SCALE 0.125f   // 1/sqrt(64)
#define ASTRIDE 68         // padded LDS row stride (floats): 16B-aligned, bank-conflict-free

typedef __attribute__((ext_vector_type(2))) float v2f;
typedef __attribute__((ext_vector_type(8))) float v8f;

// ---------------- float atomic max via CAS (safe everywhere) ----------------
__device__ __forceinline__ void atomicMaxF(float* addr, float val) {
  unsigned int* ua = (unsigned int*)addr;
  unsigned int cur = __float_as_uint(*addr);
  while (__uint_as_float(cur) < val) {
    unsigned int assumed = cur;
    cur = atomicCAS(ua, assumed, __float_as_uint(val));
    if (cur == assumed) break;
  }
}

// ---------------------------------------------------------------------------
// Fused Q/K/V/Skip projection with LDS-staged A tile.
// Block = 128 threads (4 waves). The block stages a 16x64 chunk of X into LDS
// once; wave w computes the 16x16 col-tile w of all four outputs:
//   q|k|v|agg[16x64] = X[16xK] @ W*[Kx64] + b*
// WMMA f32 16x16x4 layouts (wave32):
//   A: lane&15 = M, lane>>4 selects K pair {0,1} vs {2,3}; v2f = (K, K+1)
//   B: lane&15 = N, lane>>4 selects K pair;                v2f = (K, K+1)
//   C: lane&15 = N, VGPR j holds M = j + 8*(lane>>4)
// ---------------------------------------------------------------------------
__global__ __launch_bounds__(128) void gnn_qkvs_wmma(
    const float* __restrict__ X, int K,
    const float* __restrict__ Wq, const float* __restrict__ bq,
    const float* __restrict__ Wk, const float* __restrict__ bk,
    const float* __restrict__ Wv, const float* __restrict__ bv,
    const float* __restrict__ Ws, const float* __restrict__ bs,
    float* __restrict__ q, float* __restrict__ k, float* __restrict__ v,
    float* __restrict__ agg)
{
  __shared__ float As[16 * ASTRIDE];

  const int tid   = threadIdx.x;     // 0..127
  const int wave  = tid >> 5;        // col tile index 0..3
  const int lane  = tid & 31;
  const int khalf = lane >> 4;
  const int nl    = lane & 15;
  const int row0  = blockIdx.x << 4;
  const int col0  = wave << 4;
  const int bcol  = col0 + nl;

  v8f cq = {}; v8f ck = {}; v8f cv = {}; v8f cs = {};

  for (int kc = 0; kc < K; kc += 64) {
    __syncthreads();
    // --- cooperative stage: 16 rows x 64 floats = 256 float4, 2 per thread ---
    #pragma unroll
    for (int i = 0; i < 2; ++i) {
      const int idx = tid + 128 * i;         // 0..255
      const int r   = idx >> 4;              // row 0..15
      const int c4  = idx & 15;              // float4 col 0..15
      const float4 val =
          *(const float4*)(X + (size_t)(row0 + r) * K + kc + c4 * 4);
      *(float4*)(&As[r * ASTRIDE + c4 * 4]) = val;
    }
    __syncthreads();

    // --- 16 WMMA K-steps over the staged chunk ---
    #pragma unroll 4
    for (int kk = 0; kk < 64; kk += 4) {
      const float* ap = &As[nl * ASTRIDE + kk + 2 * khalf];
      v2f a; a[0] = ap[0]; a[1] = ap[1];
      const int r0 = (kc + kk + 2 * khalf) * HD + bcol;
      v2f fq; fq[0] = Wq[r0]; fq[1] = Wq[r0 + HD];
      v2f fk; fk[0] = Wk[r0]; fk[1] = Wk[r0 + HD];
      v2f fv; fv[0] = Wv[r0]; fv[1] = Wv[r0 + HD];
      v2f fs; fs[0] = Ws[r0]; fs[1] = Ws[r0 + HD];
      cq = __builtin_amdgcn_wmma_f32_16x16x4_f32(false, a, false, fq, (short)0, cq, false, false);
      ck = __builtin_amdgcn_wmma_f32_16x16x4_f32(false, a, false, fk, (short)0, ck, false, false);
      cv = __builtin_amdgcn_wmma_f32_16x16x4_f32(false, a, false, fv, (short)0, cv, false, false);
      cs = __builtin_amdgcn_wmma_f32_16x16x4_f32(false, a, false, fs, (short)0, cs, false, false);
    }
  }

  const float biasq = bq[bcol], biask = bk[bcol];
  const float biasv = bv[bcol], biass = bs[bcol];
  #pragma unroll
  for (int j = 0; j < 8; ++j) {
    const size_t idx = (size_t)(row0 + 8 * khalf + j) * HD + bcol;
    q[idx]   = cq[j] + biasq;
    k[idx]   = ck[j] + biask;
    v[idx]   = cv[j] + biasv;
    agg[idx] = cs[j] + biass;   // skip/root term; attention aggregated on top
  }
}

// out = BN(ReLU(A[M,64] @ W[64,64] + bias)) -- eval-mode BN with running stats
__global__ __launch_bounds__(32) void gnn_lin_relu_bn_wmma(
    const float* __restrict__ A, const float* __restrict__ W,
    const float* __restrict__ bias,
    const float* __restrict__ bng, const float* __restrict__ bnb,
    const float* __restrict__ bnrm, const float* __restrict__ bnrv,
    float* __restrict__ out)
{
  const int lane  = threadIdx.x;
  const int khalf = lane >> 4;
  const int nl    = lane & 15;
  const int row0  = blockIdx.x << 4;
  const int col0  = blockIdx.y << 4;

  v8f c = {};
  const float* arow = A + (size_t)(row0 + nl) * HD + 2 * khalf;
  const int bcol = col0 + nl;

  #pragma unroll 4
  for (int k0 = 0; k0 < HD; k0 += 4) {
    v2f a; a[0] = arow[k0]; a[1] = arow[k0 + 1];
    const int r0 = (k0 + 2 * khalf) * HD + bcol;
    v2f b; b[0] = W[r0]; b[1] = W[r0 + HD];
    c = __builtin_amdgcn_wmma_f32_16x16x4_f32(false, a, false, b, (short)0, c, false, false);
  }

  const float bsv   = bias[bcol];
  const float scale = rsqrtf(bnrv[bcol] + BN_EPS) * bng[bcol];
  const float shift = bnb[bcol] - bnrm[bcol] * scale;
  #pragma unroll
  for (int j = 0; j < 8; ++j) {
    float val = fmaxf(c[j] + bsv, 0.0f);
    out[(size_t)(row0 + 8 * khalf + j) * HD + bcol] = val * scale + shift;
  }
}

// ------------------------------- edge phase --------------------------------
__global__ void gnn_init_ms(float* __restrict__ m, float* __restrict__ s) {
  int i = blockIdx.x * blockDim.x + threadIdx.x;
  if (i < NNODES) { m[i] = -3.0e38f; s[i] = 0.0f; }
}

__global__ void gnn_edge_logits(const int* __restrict__ ei,
                                const float* __restrict__ q,
                                const float* __restrict__ k,
                                float* __restrict__ logits,
                                float* __restrict__ m)
{
  int e = blockIdx.x * blockDim.x + threadIdx.x;
  if (e >= NEDGES) return;
  const int src = ei[e];
  const int dst = ei[NEDGES + e];
  const float4* qr = (const float4*)(q + (size_t)dst * HD);
  const float4* kr = (const float4*)(k + (size_t)src * HD);
  float acc = 0.0f;
  #pragma unroll
  for (int i = 0; i < 16; ++i) {
    float4 aq = qr[i], ak = kr[i];
    acc += aq.x * ak.x + aq.y * ak.y + aq.z * ak.z + aq.w * ak.w;
  }
  acc *= ATT_SCALE;
  logits[e] = acc;
  atomicMaxF(&m[dst], acc);
}

__global__ void gnn_edge_expsum(const int* __restrict__ ei,
                                float* __restrict__ logits,
                                const float* __restrict__ m,
                                float* __restrict__ s)
{
  int e = blockIdx.x * blockDim.x + threadIdx.x;
  if (e >= NEDGES) return;
  const int dst = ei[NEDGES + e];
  const float ev = expf(logits[e] - m[dst]);
  logits[e] = ev;               // reuse buffer: now holds exp(logit - max)
  atomicAdd(&s[dst], ev);
}

__global__ void gnn_edge_scatter(const int* __restrict__ ei,
                                 const float* __restrict__ logits,
                                 const float* __restrict__ s,
                                 const float* __restrict__ v,
                                 float* __restrict__ agg)
{
  long long t = (long long)blockIdx.x * blockDim.x + threadIdx.x;
  if (t >= (long long)NEDGES * HD) return;
  const int e = (int)(t >> 6);
  const int h = (int)(t & 63);
  const int src = ei[e];
  const int dst = ei[NEDGES + e];
  const float alpha = logits[e] / s[dst];
  atomicAdd(&agg[(size_t)dst * HD + h], alpha * v[(size_t)src * HD + h]);
}

// ------------------------------- pooling -----------------------------------
__global__ void gnn_init_pool(float* __restrict__ pool, float* __restrict__ cnt) {
  int i = blockIdx.x * blockDim.x + threadIdx.x;
  if (i < NGRAPH * HD) pool[i] = 0.0f;
  if (i < NGRAPH)      cnt[i]  = 0.0f;
}

__global__ void gnn_pool_accum(const float* __restrict__ h,
                               const int* __restrict__ batch,
                               float* __restrict__ pool,
                               float* __restrict__ cnt)
{
  int t = blockIdx.x * blockDim.x + threadIdx.x;
  if (t >= NNODES * HD) return;
  const int n  = t >> 6;
  const int hh = t & 63;
  const int g  = batch[n];
  atomicAdd(&pool[g * HD + hh], h[t]);
  if (hh == 0) atomicAdd(&cnt[g], 1.0f);
}

__global__ void gnn_pool_div(const float* __restrict__ pool,
                             const float* __restrict__ cnt,
                             float* __restrict__ out)
{
  int t = blockIdx.x * blockDim.x + threadIdx.x;
  if (t >= NGRAPH * HD) return;
  out[t] = pool[t] / fmaxf(cnt[t >> 6], 1.0f);
}

// ---------------------------- MLP head + softmax ---------------------------
__global__ void gnn_head(const float* __restrict__ pooled,
                         const float* __restrict__ W1, const float* __restrict__ b1,
                         const float* __restrict__ W2, const float* __restrict__ b2,
                         const float* __restrict__ W3, const float* __restrict__ b3,
                         float* __restrict__ out2)
{
  int g = blockIdx.x * blockDim.x + threadIdx.x;
  if (g >= NGRAPH) return;
  float h1[64];
  for (int o = 0; o < 64; ++o) {
    float acc = b1[o];
    for (int i = 0; i < 64; ++i) acc += pooled[g * 64 + i] * W1[i * 64 + o];
    h1[o] = fmaxf(acc, 0.0f);
  }
  float h2[32];
  for (int o = 0; o < 32; ++o) {
    float acc = b2[o];
    for (int i = 0; i < 64; ++i) acc += h1[i] * W2[i * 32 + o];
    h2[o] = fmaxf(acc, 0.0f);
  }
  float l0 = b3[0], l1 = b3[1];
  for (int i = 0; i < 32; ++i) { l0 += h2[i] * W3[i * 2]; l1 += h2[i] * W3[i * 2 + 1]; }
  const float mx = fmaxf(l0, l1);
  const float e0 = expf(l0 - mx), e1 = expf(l1 - mx);
  const float inv = 1.0f / (e0 + e1);
  out2[g * 2 + 0] = e0 * inv;
  out2[g * 2 + 1] = e1 * inv;
}

// ---------------------------------------------------------------------------
extern "C" void kernel_launch(void* const* d_in, const int* in_sizes, int n_in,
                              void* d_out, int out_size, void* d_ws, size_t ws_size,
                              hipStream_t stream) {
  (void)in_sizes; (void)n_in; (void)out_size; (void)ws_size;

  // Inputs in setup_inputs() insertion order (recursively):
  // 0:x 1:edge_index 2:edge_attr 3:batch
  // per layer l (base = 4 + 14*l): Wq,bq,Wk,bk,Wv,bv,Ws,bs, linW,linb, g,b,rm,rv
  // 60..65: head1 W,b  head2 W,b  head3 W,b
  const float* x     = (const float*)d_in[0];
  const int*   ei    = (const int*)d_in[1];
  const int*   batch = (const int*)d_in[3];

  float* ws   = (float*)d_ws;
  float* q    = ws;
  float* kbuf = q    + (size_t)NNODES * HD;
  float* vbuf = kbuf + (size_t)NNODES * HD;
  float* agg  = vbuf + (size_t)NNODES * HD;
  float* hbuf = agg  + (size_t)NNODES * HD;
  float* lgt  = hbuf + (size_t)NNODES * HD;
  float* mbuf = lgt  + (size_t)NEDGES;
  float* sbuf = mbuf + (size_t)NNODES;
  float* pool = sbuf + (size_t)NNODES;
  float* cnt  = pool + (size_t)NGRAPH * HD;

  const dim3 qkvsGrid(NNODES / 16);          // 4-wave blocks, each covers 64 cols
  const dim3 linGrid(NNODES / 16, HD / 16);  // 1-wave blocks, 16x16 tiles
  const int nBlk  = (NNODES + 255) / 256;
  const int eBlk  = (NEDGES + 255) / 256;
  const int ehBlk = (int)(((long long)NEDGES * HD + 255) / 256);
  const int nhBlk = (NNODES * HD + 255) / 256;
  const int pBlk  = (NGRAPH * HD + 255) / 256;

  for (int l = 0; l < 4; ++l) {
    const int base = 4 + 14 * l;
    const float* Wq  = (const float*)d_in[base + 0];
    const float* bq  = (const float*)d_in[base + 1];
    const float* Wk  = (const float*)d_in[base + 2];
    const float* bk  = (const float*)d_in[base + 3];
    const float* Wv  = (const float*)d_in[base + 4];
    const float* bv  = (const float*)d_in[base + 5];
    const float* Wsk = (const float*)d_in[base + 6];
    const float* bsk = (const float*)d_in[base + 7];
    const float* Wl  = (const float*)d_in[base + 8];
    const float* bl  = (const float*)d_in[base + 9];
    const float* bg  = (const float*)d_in[base + 10];
    const float* bb  = (const float*)d_in[base + 11];
    const float* rm  = (const float*)d_in[base + 12];
    const float* rv  = (const float*)d_in[base + 13];

    const float* Xin = (l == 0) ? x : hbuf;
    const int    K   = (l == 0) ? FIN : HD;

    gnn_qkvs_wmma<<<qkvsGrid, 128, 0, stream>>>(Xin, K, Wq, bq, Wk, bk, Wv, bv,
                                                Wsk, bsk, q, kbuf, vbuf, agg);
    gnn_init_ms<<<nBlk, 256, 0, stream>>>(mbuf, sbuf);
    gnn_edge_logits<<<eBlk, 256, 0, stream>>>(ei, q, kbuf, lgt, mbuf);
    gnn_edge_expsum<<<eBlk, 256, 0, stream>>>(ei, lgt, mbuf, sbuf);
    gnn_edge_scatter<<<ehBlk, 256, 0, stream>>>(ei, lgt, sbuf, vbuf, agg);
    gnn_lin_relu_bn_wmma<<<linGrid, 32, 0, stream>>>(agg, Wl, bl, bg, bb, rm, rv, hbuf);
  }

  gnn_init_pool<<<pBlk, 256, 0, stream>>>(pool, cnt);
  gnn_pool_accum<<<nhBlk, 256, 0, stream>>>(hbuf, batch, pool, cnt);
  gnn_pool_div<<<pBlk, 256, 0, stream>>>(pool, cnt, (float*)d_out);

  gnn_head<<<1, 128, 0, stream>>>((const float*)d_out,
                                  (const float*)d_in[60], (const float*)d_in[61],
                                  (const float*)d_in[62], (const float*)d_in[63],
                                  (const float*)d_in[64], (const float*)d_in[65],
                                  (float*)d_out + NGRAPH * HD);
}